// QModel_65481071406360
// MI455X (gfx1250) — compile-verified
//
#include <hip/hip_runtime.h>

#define QDIM   65536
#define QBATCH 256

typedef float v2f __attribute__((ext_vector_type(2)));
typedef float v8f __attribute__((ext_vector_type(8)));

// prefix-XOR from MSB over 16 bits: y_i = x15 ^ ... ^ x_i  (the full CNOT-chain permutation)
__device__ __forceinline__ unsigned prefxor16(unsigned x) {
  x ^= x >> 1; x ^= x >> 2; x ^= x >> 4; x ^= x >> 8;
  return x & 0xFFFFu;
}

// ---------------------------------------------------------------------------
// Build the per-(layer,group) 16x16 complex group unitaries: kron of four
// RZ*RY*RX 2x2 rotations. One thread per output element (2*4*16*16 = 2048).
// mats layout: [(layer*4+group)] blocks of 512 floats: [0..255]=Re, [256..511]=Im
// ---------------------------------------------------------------------------
__global__ void qc_prep(const float* __restrict__ params, float* __restrict__ mats) {
  int tid   = blockIdx.x * blockDim.x + threadIdx.x;   // 0..2047
  int layer = tid >> 10;
  int group = (tid >> 8) & 3;
  int i     = (tid >> 4) & 15;
  int j     = tid & 15;

  float accRe = 1.0f, accIm = 0.0f;
  #pragma unroll
  for (int q = 0; q < 4; ++q) {
    int w = group * 4 + q;
    const float* p = params + (layer * 16 + w) * 3;
    float cx = cosf(p[0] * 0.5f), sx = sinf(p[0] * 0.5f);
    float cy = cosf(p[1] * 0.5f), sy = sinf(p[1] * 0.5f);
    float ca = cosf(p[2] * 0.5f), sa = sinf(p[2] * 0.5f);
    int ib = (i >> (3 - q)) & 1;   // wire 4g+q is bit (3-q) of the nibble (wire 0 = MSB)
    int jb = (j >> (3 - q)) & 1;
    // Ry*Rx entry = pp + i*qq
    float pp, qq;
    if      (ib == 0 && jb == 0) { pp =  cy * cx; qq =  sy * sx; }
    else if (ib == 0)            { pp = -sy * cx; qq = -cy * sx; }
    else if (jb == 0)            { pp =  sy * cx; qq = -cy * sx; }
    else                         { pp =  cy * cx; qq = -sy * sx; }
    // Rz row phase: row0 *= e^{-ia}, row1 *= e^{+ia}
    float ur, ui;
    if (ib == 0) { ur = pp * ca + qq * sa; ui = qq * ca - pp * sa; }
    else         { ur = pp * ca - qq * sa; ui = qq * ca + pp * sa; }
    float nr = accRe * ur - accIm * ui;
    float ni = accRe * ui + accIm * ur;
    accRe = nr; accIm = ni;
  }
  int mbase = (layer * 4 + group) * 512;
  mats[mbase +       i * 16 + j] = accRe;
  mats[mbase + 256 + i * 16 + j] = accIm;
}

// ---------------------------------------------------------------------------
// Per-batch squared norm (normalization is deferred to the output head).
// ---------------------------------------------------------------------------
__global__ __launch_bounds__(256) void qc_norm(const float* __restrict__ re,
                                               const float* __restrict__ im,
                                               float* __restrict__ norm2) {
  __shared__ float red[256];
  int b = blockIdx.x;
  const float* r  = re + (size_t)b * QDIM;
  const float* ii = im + (size_t)b * QDIM;
  float acc = 0.0f;
  for (int it = 0; it < 256; ++it) {
    int x = it * 256 + (int)threadIdx.x;
    float a = r[x], c = ii[x];
    acc += a * a + c * c;
  }
  red[threadIdx.x] = acc; __syncthreads();
  for (int s = 128; s > 0; s >>= 1) {
    if ((int)threadIdx.x < s) red[threadIdx.x] += red[threadIdx.x + s];
    __syncthreads();
  }
  if (threadIdx.x == 0) norm2[b] = red[0];
}

// ---------------------------------------------------------------------------
// Apply one group's 16x16 complex unitary along its nibble axis via f32 WMMA.
// One wave = one 16x16 tile per loop iter; tiles partition the state, so
// in-place (src==dst) is race-free. permLoads folds the inter-layer CNOT
// permutation (inverse prefix-XOR) into the load addresses.
// ---------------------------------------------------------------------------
__global__ __launch_bounds__(256) void qc_group_pass(
    const float* __restrict__ srcRe, const float* __restrict__ srcIm,
    float* __restrict__ dstRe, float* __restrict__ dstIm,
    const float* __restrict__ mats,   // 256 Re then 256 Im for this (layer,group)
    int group, int permLoads, int tilesPerWave)
{
  const int lane = (int)(threadIdx.x & 31u);
  const int n    = lane & 15;       // B/D column, A row
  const int hi   = lane >> 4;       // half-wave selector

  // A tiles: M 16x16 split into four 16x4 K-chunks.
  // A layout: lane m = lane&15 ; VGPR r holds K = kc*4 + r + 2*hi
  v2f are[4], aim[4], ain[4];
  #pragma unroll
  for (int kc = 0; kc < 4; ++kc) {
    int col = kc * 4 + 2 * hi;
    float r0 = mats[n * 16 + col];
    float r1 = mats[n * 16 + col + 1];
    float i0 = mats[256 + n * 16 + col];
    float i1 = mats[256 + n * 16 + col + 1];
    v2f tr; tr[0] = r0;  tr[1] = r1;  are[kc] = tr;
    v2f ti; ti[0] = i0;  ti[1] = i1;  aim[kc] = ti;
    v2f tn; tn[0] = -i0; tn[1] = -i1; ain[kc] = tn;
  }

  const int waveId    = (int)blockIdx.x * ((int)blockDim.x >> 5) + ((int)threadIdx.x >> 5);
  const int rowStride = (group == 0) ? 4096 : (group == 1) ? 256 : (group == 2) ? 16 : 1;
  const int colStride = (group == 3) ? 16 : 1;

  for (int t = 0; t < tilesPerWave; ++t) {
    int tile = waveId * tilesPerWave + t;       // 0..65535
    int b    = tile >> 8;
    int tt   = tile & 255;
    int baseIn;
    if      (group == 0) baseIn = tt * 16;
    else if (group == 1) baseIn = (tt >> 4) * 4096 + (tt & 15) * 16;
    else                 baseIn = tt * 256;
    int base = b * QDIM + baseIn;

    // B tiles: state 16(K=group nibble) x 16(N=positions), split into 4x16 K-chunks.
    v2f bre[4], bim[4];
    #pragma unroll
    for (int kc = 0; kc < 4; ++kc) {
      float vr[2], vi[2];
      #pragma unroll
      for (int r = 0; r < 2; ++r) {
        int k   = kc * 4 + 2 * hi + r;
        unsigned idx = (unsigned)(base + k * rowStride + n * colStride);
        if (permLoads) {
          unsigned low = idx & 65535u;
          low = (low ^ (low >> 1)) & 65535u;   // inverse of prefix-XOR permutation
          idx = (idx & ~65535u) | low;
        }
        vr[r] = srcRe[idx];
        vi[r] = srcIm[idx];
      }
      v2f tr; tr[0] = vr[0]; tr[1] = vr[1]; bre[kc] = tr;
      v2f ti; ti[0] = vi[0]; ti[1] = vi[1]; bim[kc] = ti;
    }

    // Complex 16x16x16 matmul = 16 x V_WMMA_F32_16X16X4_F32
    v8f cre = {};
    v8f cim = {};
    #pragma unroll
    for (int kc = 0; kc < 4; ++kc) {
      cre = __builtin_amdgcn_wmma_f32_16x16x4_f32(false, are[kc], false, bre[kc], (short)0, cre, false, false);
      cre = __builtin_amdgcn_wmma_f32_16x16x4_f32(false, ain[kc], false, bim[kc], (short)0, cre, false, false);
      cim = __builtin_amdgcn_wmma_f32_16x16x4_f32(false, aim[kc], false, bre[kc], (short)0, cim, false, false);
      cim = __builtin_amdgcn_wmma_f32_16x16x4_f32(false, are[kc], false, bim[kc], (short)0, cim, false, false);
    }

    // D layout: VGPR mv holds row M = mv + 8*hi, column = n. Stores never permuted.
    #pragma unroll
    for (int mv = 0; mv < 8; ++mv) {
      int m = mv + 8 * hi;
      int idx = base + m * rowStride + n * colStride;
      dstRe[idx] = cre[mv];
      dstIm[idx] = cim[mv];
    }
  }
}

// ---------------------------------------------------------------------------
// Pauli-Z expvals (with the final CNOT permutation folded into the sign
// lookup via prefix-XOR), linear head, deferred 1/norm^2.
// ---------------------------------------------------------------------------
__global__ __launch_bounds__(256) void qc_out(const float* __restrict__ re,
                                              const float* __restrict__ im,
                                              const float* __restrict__ head_w,
                                              const float* __restrict__ head_b,
                                              const float* __restrict__ norm2,
                                              float* __restrict__ out) {
  __shared__ float tbl[64];    // per-nibble head_w-weighted sign sums
  __shared__ float red[256];
  int b = blockIdx.x;
  if (threadIdx.x < 64) {
    int g = (int)threadIdx.x >> 4, v = (int)threadIdx.x & 15;
    float s = 0.0f;
    #pragma unroll
    for (int q = 0; q < 4; ++q) {
      float sign = ((v >> (3 - q)) & 1) ? -1.0f : 1.0f;
      s += head_w[g * 4 + q] * sign;
    }
    tbl[threadIdx.x] = s;
  }
  __syncthreads();
  const float* r  = re + (size_t)b * QDIM;
  const float* ii = im + (size_t)b * QDIM;
  float acc = 0.0f;
  for (int it = 0; it < 256; ++it) {
    int x = it * 256 + (int)threadIdx.x;
    float a = r[x], c = ii[x];
    float p = a * a + c * c;
    unsigned y = prefxor16((unsigned)x);     // final-layer CNOT permutation
    acc += p * (tbl[y >> 12] + tbl[16 + ((y >> 8) & 15)] +
                tbl[32 + ((y >> 4) & 15)] + tbl[48 + (y & 15)]);
  }
  red[threadIdx.x] = acc; __syncthreads();
  for (int s = 128; s > 0; s >>= 1) {
    if ((int)threadIdx.x < s) red[threadIdx.x] += red[threadIdx.x + s];
    __syncthreads();
  }
  if (threadIdx.x == 0) out[b] = red[0] / norm2[b] + head_b[0];
}

// ---------------------------------------------------------------------------
extern "C" void kernel_launch(void* const* d_in, const int* in_sizes, int n_in,
                              void* d_out, int out_size, void* d_ws, size_t ws_size,
                              hipStream_t stream) {
  (void)in_sizes; (void)n_in; (void)out_size; (void)ws_size;
  const float* state_re = (const float*)d_in[0];
  const float* state_im = (const float*)d_in[1];
  const float* params   = (const float*)d_in[2];
  const float* head_w   = (const float*)d_in[3];
  const float* head_b   = (const float*)d_in[4];
  float* out = (float*)d_out;

  float* ws    = (float*)d_ws;
  float* norm2 = ws;                 // 256 floats
  float* mats  = ws + 256;           // 2*4*512 = 4096 floats
  float* S0re  = ws + 8192;
  float* S0im  = S0re + (size_t)QBATCH * QDIM;
  float* S1re  = S0im + (size_t)QBATCH * QDIM;
  float* S1im  = S1re + (size_t)QBATCH * QDIM;

  qc_prep<<<8, 256, 0, stream>>>(params, mats);
  qc_norm<<<256, 256, 0, stream>>>(state_re, state_im, norm2);

  const int blocks = 1024, tpb = 256;
  const int waves  = blocks * (tpb / 32);       // 8192
  const int tilesPerWave = 65536 / waves;       // 8, exact -> uniform EXEC for WMMA

  auto M = [&](int layer, int group) { return mats + (layer * 4 + group) * 512; };

  // Layer 0: four group passes (first one reads inputs, rest in-place).
  qc_group_pass<<<blocks, tpb, 0, stream>>>(state_re, state_im, S0re, S0im, M(0, 0), 0, 0, tilesPerWave);
  qc_group_pass<<<blocks, tpb, 0, stream>>>(S0re, S0im, S0re, S0im, M(0, 1), 1, 0, tilesPerWave);
  qc_group_pass<<<blocks, tpb, 0, stream>>>(S0re, S0im, S0re, S0im, M(0, 2), 2, 0, tilesPerWave);
  qc_group_pass<<<blocks, tpb, 0, stream>>>(S0re, S0im, S0re, S0im, M(0, 3), 3, 0, tilesPerWave);
  // Layer 1: first pass folds the layer-0 CNOT-chain permutation into loads.
  qc_group_pass<<<blocks, tpb, 0, stream>>>(S0re, S0im, S1re, S1im, M(1, 0), 0, 1, tilesPerWave);
  qc_group_pass<<<blocks, tpb, 0, stream>>>(S1re, S1im, S1re, S1im, M(1, 1), 1, 0, tilesPerWave);
  qc_group_pass<<<blocks, tpb, 0, stream>>>(S1re, S1im, S1re, S1im, M(1, 2), 2, 0, tilesPerWave);
  qc_group_pass<<<blocks, tpb, 0, stream>>>(S1re, S1im, S1re, S1im, M(1, 3), 3, 0, tilesPerWave);

  qc_out<<<256, 256, 0, stream>>>(S1re, S1im, head_w, head_b, norm2, out);
}